// LabelPropagation_8452495638815
// MI455X (gfx1250) — compile-verified
//
#include <hip/hip_runtime.h>
#include <hip/hip_bf16.h>

typedef __attribute__((ext_vector_type(2))) float v2f;
typedef __attribute__((ext_vector_type(8))) float v8f;
typedef __attribute__((ext_vector_type(4))) unsigned v4u;
typedef __attribute__((ext_vector_type(8))) unsigned v8u;

#define N_ALL 4096
#define N_SUP 1024
#define DIM 1024
#define NWORDS 128            // 4096 bits / 32
#define EPSF 2.2204460492503131e-16f
#define ALPHAF 0.99f
#define SOLVE_ITERS 2000

// ---------------------------------------------------------------- row norms
__global__ __launch_bounds__(256) void sqnorm_kernel(
    const float* __restrict__ lab, const float* __restrict__ unlab,
    float* __restrict__ sq) {
  const int lane = threadIdx.x & 31, wid = threadIdx.x >> 5;
  const int row = blockIdx.x * 8 + wid;
  const float* src = (row < N_SUP) ? (lab + (size_t)row * DIM)
                                   : (unlab + (size_t)(row - N_SUP) * DIM);
  float s = 0.0f;
  for (int c = lane; c < DIM; c += 32) { float v = src[c]; s += v * v; }
  for (int off = 16; off > 0; off >>= 1) s += __shfl_down(s, off, 32);
  if (lane == 0) sq[row] = s;
}

// ---------------------------------------------------------------- zero words
__global__ void zero_u32_kernel(unsigned* __restrict__ p, int n) {
  int i = blockIdx.x * blockDim.x + threadIdx.x;
  if (i < n) p[i] = 0u;
}

// ---------------------------------------------------------------- TDM: DMA a rows x 32-float tile into LDS
// LDS layout: row stride 34 floats (pad 2 DWORDs after every 32 DWORDs -> bank-conflict-free)
__device__ inline void tdm_load_tile(unsigned lds_off, const float* gptr, unsigned rows) {
  unsigned long long ga = (unsigned long long)(size_t)gptr;
  v4u g0;
  g0.x = 1u;                                                  // count=1, user descriptor
  g0.y = lds_off;                                             // lds_addr (bytes)
  g0.z = (unsigned)(ga & 0xffffffffu);                        // global_addr[31:0]
  g0.w = (unsigned)((ga >> 32) & 0x01ffffffu) | (2u << 30);   // global_addr[56:32] | type=2
  v8u g1;
  g1.s0 = (2u << 16)    // data_size = 4B
        | (1u << 20)    // pad_enable
        | (4u << 22)    // pad_interval: every 32 DWORDs
        | (1u << 25);   // pad_amount: 2 DWORDs
  g1.s1 = (unsigned)(DIM & 0xffffu) << 16;                    // tensor_dim0[15:0]=1024
  g1.s2 = ((unsigned)DIM >> 16) | ((rows & 0xffffu) << 16);   // dim0[31:16] | tensor_dim1[15:0]
  g1.s3 = ((rows >> 16) & 0xffffu) | (32u << 16);             // dim1[31:16] | tile_dim0=32
  g1.s4 = rows & 0xffffu;                                     // tile_dim1=rows, tile_dim2=0
  g1.s5 = (unsigned)DIM;                                      // tensor_dim0_stride[31:0]=1024
  g1.s6 = 0u;                                                 // stride hi / dim1_stride lo
  g1.s7 = 0u;
  // 2-operand form: VADDR2/VADDR3 = NULL -> 2D tensor descriptor
  asm volatile("tensor_load_to_lds %0, %1" :: "s"(g0), "s"(g1) : "memory");
}

// ---------------------------------------------------------------- W = exp Gram (WMMA f32 + TDM double buffer)
#define LDA 34
__global__ __launch_bounds__(256) void gram_w_kernel(
    const float* __restrict__ lab, const float* __restrict__ unlab,
    const float* __restrict__ sq, float* __restrict__ Wm) {
  __shared__ float As[2][128 * LDA];
  __shared__ float Bs[2][64 * LDA];
  const int tid = threadIdx.x;
  const int lane = tid & 31, wid = tid >> 5;
  const int half = lane >> 4, lrow = lane & 15;
  const int i0 = blockIdx.y * 128, j0 = blockIdx.x * 64;
  const int wm = (wid & 3) * 32, wn = (wid >> 2) * 32;

  // row blocks never straddle the labeled/unlabeled boundary (128 | 1024, 64 | 1024)
  const float* arow = (i0 < N_SUP) ? (lab + (size_t)i0 * DIM)
                                   : (unlab + (size_t)(i0 - N_SUP) * DIM);
  const float* brow = (j0 < N_SUP) ? (lab + (size_t)j0 * DIM)
                                   : (unlab + (size_t)(j0 - N_SUP) * DIM);
  const unsigned asoff[2] = {(unsigned)(size_t)&As[0][0], (unsigned)(size_t)&As[1][0]};
  const unsigned bsoff[2] = {(unsigned)(size_t)&Bs[0][0], (unsigned)(size_t)&Bs[1][0]};

  const v8f vzero = {0.f, 0.f, 0.f, 0.f, 0.f, 0.f, 0.f, 0.f};
  v8f acc[2][2];
  acc[0][0] = vzero; acc[0][1] = vzero; acc[1][0] = vzero; acc[1][1] = vzero;

  if (wid == 0) {                       // wave 0 drives the TDM pipeline
    tdm_load_tile(asoff[0], arow, 128);
    tdm_load_tile(bsoff[0], brow, 64);
  }

  for (int t = 0; t < DIM / 32; ++t) {
    const int buf = t & 1;
    if (wid == 0) {
      if (t < DIM / 32 - 1) {
        tdm_load_tile(asoff[buf ^ 1], arow + (t + 1) * 32, 128);
        tdm_load_tile(bsoff[buf ^ 1], brow + (t + 1) * 32, 64);
        __builtin_amdgcn_s_wait_tensorcnt(2);   // chunk t landed (in-order), t+1 in flight
      } else {
        __builtin_amdgcn_s_wait_tensorcnt(0);
      }
    }
    __syncthreads();
    const float* Ab = As[buf];
    const float* Bb = Bs[buf];
#pragma unroll
    for (int k = 0; k < 32; k += 4) {
      // A 16x4 f32: lanes 0-15 hold K=k..k+1 of row lrow; lanes 16-31 hold K=k+2..k+3
      v2f a0 = *reinterpret_cast<const v2f*>(&Ab[(wm +      lrow) * LDA + k + half * 2]);
      v2f a1 = *reinterpret_cast<const v2f*>(&Ab[(wm + 16 + lrow) * LDA + k + half * 2]);
      v2f b0 = *reinterpret_cast<const v2f*>(&Bb[(wn +      lrow) * LDA + k + half * 2]);
      v2f b1 = *reinterpret_cast<const v2f*>(&Bb[(wn + 16 + lrow) * LDA + k + half * 2]);
      acc[0][0] = __builtin_amdgcn_wmma_f32_16x16x4_f32(false, a0, false, b0, (short)0, acc[0][0], false, false);
      acc[0][1] = __builtin_amdgcn_wmma_f32_16x16x4_f32(false, a0, false, b1, (short)0, acc[0][1], false, false);
      acc[1][0] = __builtin_amdgcn_wmma_f32_16x16x4_f32(false, a1, false, b0, (short)0, acc[1][0], false, false);
      acc[1][1] = __builtin_amdgcn_wmma_f32_16x16x4_f32(false, a1, false, b1, (short)0, acc[1][1], false, false);
    }
    __syncthreads();   // all waves done reading buf before TDM refills it
  }

  // epilogue: G -> W = exp(-dist/2), zero diagonal
#pragma unroll
  for (int sm = 0; sm < 2; sm++) {
#pragma unroll
    for (int sn = 0; sn < 2; sn++) {
      int n = j0 + wn + sn * 16 + lrow;
      float sqn = sq[n];
#pragma unroll
      for (int r = 0; r < 8; r++) {
        int m = i0 + wm + sm * 16 + half * 8 + r;  // C/D layout: lanes>=16 -> M+8
        float g = acc[sm][sn][r];
        float dist = (sq[m] + sqn - 2.0f * g) * (1.0f / (float)DIM);
        float w = __expf(-0.5f * dist);            // SIGMA = 1
        if (m == n) w = 0.0f;
        Wm[(size_t)m * N_ALL + n] = w;
      }
    }
  }
}

// ---------------------------------------------------------------- per-row top-4 -> symmetric mask bits
__device__ inline void tk_insert(float v, int idx, float tv[4], int ti[4]) {
  if (v > tv[3]) {
    tv[3] = v; ti[3] = idx;
#pragma unroll
    for (int s = 3; s > 0; --s) {
      if (tv[s] > tv[s - 1]) {
        float fv = tv[s]; tv[s] = tv[s - 1]; tv[s - 1] = fv;
        int fi = ti[s]; ti[s] = ti[s - 1]; ti[s - 1] = fi;
      }
    }
  }
}

__global__ __launch_bounds__(256) void topk_mask_kernel(
    const float* __restrict__ Wm, unsigned* __restrict__ maskbits) {
  __shared__ float sv[1024];
  __shared__ int   si[1024];
  const int tid = threadIdx.x;
  const int row = blockIdx.x;
  float tv[4] = {-1e30f, -1e30f, -1e30f, -1e30f};
  int   ti[4] = {0, 0, 0, 0};
  const float* Wrow = Wm + (size_t)row * N_ALL;
  for (int j = tid; j < N_ALL; j += 256) tk_insert(Wrow[j], j, tv, ti);
#pragma unroll
  for (int q = 0; q < 4; q++) { sv[tid * 4 + q] = tv[q]; si[tid * 4 + q] = ti[q]; }
  __syncthreads();
  for (int s = 128; s >= 1; s >>= 1) {
    if (tid < s) {
      float mv[4]; int mi[4];
#pragma unroll
      for (int q = 0; q < 4; q++) { mv[q] = sv[tid * 4 + q]; mi[q] = si[tid * 4 + q]; }
#pragma unroll
      for (int q = 0; q < 4; q++) tk_insert(sv[(tid + s) * 4 + q], si[(tid + s) * 4 + q], mv, mi);
#pragma unroll
      for (int q = 0; q < 4; q++) { sv[tid * 4 + q] = mv[q]; si[tid * 4 + q] = mi[q]; }
    }
    __syncthreads();
  }
  if (tid < 4) {
    int c = si[tid];
    atomicOr(&maskbits[(size_t)row * NWORDS + (c >> 5)], 1u << (c & 31));
    atomicOr(&maskbits[(size_t)c * NWORDS + (row >> 5)], 1u << (row & 31));
  }
}

// ---------------------------------------------------------------- degree + nnz count (wave per row)
__global__ __launch_bounds__(256) void degree_kernel(
    const float* __restrict__ Wm, const unsigned* __restrict__ maskbits,
    float* __restrict__ Dg, unsigned* __restrict__ cnt) {
  const int lane = threadIdx.x & 31, wid = threadIdx.x >> 5;
  const int row = blockIdx.x * 8 + wid;
  const unsigned* bits = maskbits + (size_t)row * NWORDS;
  const float* Wrow = Wm + (size_t)row * N_ALL;
  float d = 0.0f; unsigned c = 0;
#pragma unroll
  for (int q = 0; q < 4; q++) {
    unsigned w = bits[lane * 4 + q];
    c += (unsigned)__popc(w);
    int jb = (lane * 4 + q) * 32;
    while (w) { int b = __ffs(w) - 1; w &= w - 1; d += Wrow[jb + b]; }
  }
  for (int off = 16; off > 0; off >>= 1) {
    d += __shfl_down(d, off, 32);
    c += __shfl_down(c, off, 32);
  }
  if (lane == 0) { Dg[row] = d; cnt[row] = c; }
}

// ---------------------------------------------------------------- D^{-1/2}
__global__ void dinv_kernel(const float* __restrict__ Dg, float* __restrict__ dinv) {
  int i = blockIdx.x * blockDim.x + threadIdx.x;
  if (i < N_ALL) dinv[i] = sqrtf(1.0f / (Dg[i] + EPSF));
}

// ---------------------------------------------------------------- exclusive scan -> row_ptr (1 block, 1024 thr)
__global__ __launch_bounds__(1024) void scan_kernel(
    const unsigned* __restrict__ cnt, unsigned* __restrict__ rowptr) {
  __shared__ unsigned part[1024];
  const int tid = threadIdx.x;
  const int b4 = tid * 4;
  unsigned a0 = cnt[b4], a1 = cnt[b4 + 1], a2 = cnt[b4 + 2], a3 = cnt[b4 + 3];
  unsigned s = a0 + a1 + a2 + a3;
  part[tid] = s;
  __syncthreads();
  for (int off = 1; off < 1024; off <<= 1) {
    unsigned v = (tid >= off) ? part[tid - off] : 0u;
    __syncthreads();
    part[tid] += v;
    __syncthreads();
  }
  unsigned excl = part[tid] - s;
  rowptr[b4] = excl;
  rowptr[b4 + 1] = excl + a0;
  rowptr[b4 + 2] = excl + a0 + a1;
  rowptr[b4 + 3] = excl + a0 + a1 + a2;
  if (tid == 1023) rowptr[N_ALL] = part[1023];
}

// ---------------------------------------------------------------- CSR fill: val = alpha * w * d_i * d_j
__global__ __launch_bounds__(256) void csr_fill_kernel(
    const float* __restrict__ Wm, const unsigned* __restrict__ maskbits,
    const unsigned* __restrict__ rowptr, const float* __restrict__ dinv,
    int* __restrict__ ccol, float* __restrict__ cval) {
  const int lane = threadIdx.x & 31, wid = threadIdx.x >> 5;
  const int row = blockIdx.x * 8 + wid;
  const unsigned* bits = maskbits + (size_t)row * NWORDS;
  unsigned wv[4]; unsigned myc = 0;
#pragma unroll
  for (int q = 0; q < 4; q++) { wv[q] = bits[lane * 4 + q]; myc += (unsigned)__popc(wv[q]); }
  unsigned p = myc;
  for (int off = 1; off < 32; off <<= 1) {
    unsigned n = __shfl_up(p, off, 32);
    if (lane >= off) p += n;
  }
  unsigned idx = rowptr[row] + (p - myc);       // ordered exclusive offset within row
  const float di = dinv[row];
  const float* Wrow = Wm + (size_t)row * N_ALL;
#pragma unroll
  for (int q = 0; q < 4; q++) {
    unsigned w = wv[q];
    int jb = (lane * 4 + q) * 32;
    while (w) {
      int b = __ffs(w) - 1; w &= w - 1;
      int j = jb + b;
      ccol[idx] = j;
      cval[idx] = ALPHAF * Wrow[j] * di * dinv[j];
      idx++;
    }
  }
}

// ---------------------------------------------------------------- iterative solve F <- y + alpha*S*F (F in LDS)
__global__ __launch_bounds__(1024) void solve_kernel(
    const int* __restrict__ tgt, const unsigned* __restrict__ rowptr,
    const int* __restrict__ ccol, const float* __restrict__ cval,
    float* __restrict__ out) {
  __shared__ float F[N_ALL * 2];
  const int tid = threadIdx.x;
  float y0[4], y1[4];
#pragma unroll
  for (int q = 0; q < 4; q++) {
    int i = tid + q * 1024;
    float a = 0.0f, b = 0.0f;
    if (i < N_SUP) { int t = tgt[i]; a = (t == 0) ? 1.0f : 0.0f; b = (t == 1) ? 1.0f : 0.0f; }
    y0[q] = a; y1[q] = b;
    F[i * 2] = a; F[i * 2 + 1] = b;
  }
  __syncthreads();
  for (int it = 0; it < SOLVE_ITERS; ++it) {
    float nf0[4], nf1[4];
#pragma unroll
    for (int q = 0; q < 4; q++) {
      int i = tid + q * 1024;
      unsigned s = rowptr[i], e = rowptr[i + 1];
      float a = 0.0f, b = 0.0f;
      for (unsigned k = s; k < e; ++k) {
        int c = ccol[k]; float v = cval[k];
        a += v * F[c * 2];
        b += v * F[c * 2 + 1];
      }
      nf0[q] = y0[q] + a; nf1[q] = y1[q] + b;
    }
    __syncthreads();
#pragma unroll
    for (int q = 0; q < 4; q++) {
      int i = tid + q * 1024;
      F[i * 2] = nf0[q]; F[i * 2 + 1] = nf1[q];
    }
    __syncthreads();
  }
#pragma unroll
  for (int q = 0; q < 4; q++) {
    int i = tid + q * 1024;
    if (i >= N_SUP) {
      out[(size_t)(i - N_SUP) * 2]     = F[i * 2];
      out[(size_t)(i - N_SUP) * 2 + 1] = F[i * 2 + 1];
    }
  }
}

// ================================================================ launcher
extern "C" void kernel_launch(void* const* d_in, const int* in_sizes, int n_in,
                              void* d_out, int out_size, void* d_ws, size_t ws_size,
                              hipStream_t stream) {
  const float* labeled   = (const float*)d_in[0];  // [1024,1024]
  const int*   target    = (const int*)d_in[1];    // [1024]
  const float* unlabeled = (const float*)d_in[2];  // [3072,1024]
  float* out = (float*)d_out;                      // [3072,2]

  char* ws = (char*)d_ws;
  size_t off = 0;
  float* Wm = (float*)(ws + off);        off += (size_t)N_ALL * N_ALL * 4;   // 64 MB
  float* sq = (float*)(ws + off);        off += N_ALL * 4;
  unsigned* maskbits = (unsigned*)(ws + off); off += (size_t)N_ALL * NWORDS * 4; // 2 MB
  float* Dg = (float*)(ws + off);        off += N_ALL * 4;
  float* dinv = (float*)(ws + off);      off += N_ALL * 4;
  unsigned* cnt = (unsigned*)(ws + off); off += N_ALL * 4;
  unsigned* rowptr = (unsigned*)(ws + off); off += (N_ALL + 64) * 4;
  int* ccol = (int*)(ws + off);          off += 32768 * 4;                   // nnz <= 2*4*N
  float* cval = (float*)(ws + off);      off += 32768 * 4;

  // 1) row squared norms
  sqnorm_kernel<<<N_ALL / 8, 256, 0, stream>>>(labeled, unlabeled, sq);
  // 2) clear mask bitmap
  zero_u32_kernel<<<(N_ALL * NWORDS + 255) / 256, 256, 0, stream>>>(maskbits, N_ALL * NWORDS);
  // 3) W via WMMA f32 Gram, TDM double-buffered staging, fused exp epilogue
  gram_w_kernel<<<dim3(N_ALL / 64, N_ALL / 128), 256, 0, stream>>>(labeled, unlabeled, sq, Wm);
  // 4) top-4 per row -> symmetric mask bits
  topk_mask_kernel<<<N_ALL, 256, 0, stream>>>(Wm, maskbits);
  // 5) masked degrees + per-row nnz
  degree_kernel<<<N_ALL / 8, 256, 0, stream>>>(Wm, maskbits, Dg, cnt);
  // 6) D^{-1/2}
  dinv_kernel<<<(N_ALL + 255) / 256, 256, 0, stream>>>(Dg, dinv);
  // 7) row_ptr
  scan_kernel<<<1, 1024, 0, stream>>>(cnt, rowptr);
  // 8) CSR of alpha*S
  csr_fill_kernel<<<N_ALL / 8, 256, 0, stream>>>(Wm, maskbits, rowptr, dinv, ccol, cval);
  // 9) iterative label propagation solve
  solve_kernel<<<1, 1024, 0, stream>>>(target, rowptr, ccol, cval, out);
}